// GNN_64132451664423
// MI455X (gfx1250) — compile-verified
//
#include <hip/hip_runtime.h>
#include <hip/hip_bf16.h>

typedef __attribute__((ext_vector_type(2))) float v2f;
typedef __attribute__((ext_vector_type(8))) float v8f;

#define D_FEAT 128

// ---------------------------------------------------------------------------
// Degree / normalization kernels
// ---------------------------------------------------------------------------
__global__ void init_deg_kernel(unsigned* __restrict__ degc, int n) {
    int i = blockIdx.x * blockDim.x + threadIdx.x;
    if (i < n) degc[i] = 1u;   // self loop
}

__global__ void count_edges_kernel(const int* __restrict__ cols,
                                   unsigned* __restrict__ degc, int nE) {
    int e = blockIdx.x * blockDim.x + threadIdx.x;
    if (e < nE) {
        (void)__hip_atomic_fetch_add(&degc[cols[e]], 1u,
                                     __ATOMIC_RELAXED, __HIP_MEMORY_SCOPE_AGENT);
    }
}

__global__ void make_dinv_kernel(const unsigned* __restrict__ degc,
                                 float* __restrict__ dinv, int n) {
    int i = blockIdx.x * blockDim.x + threadIdx.x;
    if (i < n) {
        float d = (float)degc[i];
        dinv[i] = (d > 0.0f) ? rsqrtf(d) : 0.0f;
    }
}

// ---------------------------------------------------------------------------
// GEMM + bias via fp32 WMMA (16x16x4), A-tile staged in LDS (stride 132,
// bank-conflict-free column reads). One 16-row block; DOUT/16 waves.
// ---------------------------------------------------------------------------
template <int DOUT>
__global__ __launch_bounds__((DOUT / 16) * 32)
void gemm_bias_wmma(const float* __restrict__ A, const float* __restrict__ W,
                    const float* __restrict__ bias, float* __restrict__ C) {
    constexpr int NW  = DOUT / 16;
    constexpr int LDA = 132;                 // padded LDS row stride (floats)
    __shared__ float As[16 * LDA];

    const int tid     = threadIdx.x;
    const int rowBase = blockIdx.x * 16;

    // Stage 16x128 fp32 A tile into LDS (float4 per thread-iteration)
    const float4* A4 = (const float4*)(A + (size_t)rowBase * D_FEAT);
    for (int idx = tid; idx < 16 * (D_FEAT / 4); idx += NW * 32) {
        int r = idx >> 5, s = idx & 31;
        *(float4*)(&As[r * LDA + s * 4]) = A4[idx];
    }
    __syncthreads();

    const int wave    = tid >> 5;
    const int lane    = tid & 31;
    const int m       = lane & 15;           // row (A) / col (B,D)
    const int kg      = (lane >> 4) << 1;    // k-pair select: 0 or 2
    const int colBase = wave * 16;

    v8f c0 = {}; v8f c1 = {};                // two accumulators: break RAW chain
#pragma unroll
    for (int kk = 0; kk < D_FEAT; kk += 8) {
        v2f a, b;
        a.x = As[m * LDA + kk + kg];
        a.y = As[m * LDA + kk + kg + 1];
        b.x = W[(size_t)(kk + kg) * DOUT + colBase + m];
        b.y = W[(size_t)(kk + kg + 1) * DOUT + colBase + m];
        c0 = __builtin_amdgcn_wmma_f32_16x16x4_f32(false, a, false, b,
                                                   (short)0, c0, false, false);
        a.x = As[m * LDA + kk + 4 + kg];
        a.y = As[m * LDA + kk + 4 + kg + 1];
        b.x = W[(size_t)(kk + 4 + kg) * DOUT + colBase + m];
        b.y = W[(size_t)(kk + 5 + kg) * DOUT + colBase + m];
        c1 = __builtin_amdgcn_wmma_f32_16x16x4_f32(false, a, false, b,
                                                   (short)0, c1, false, false);
    }
    v8f c = c0 + c1;

    const float bn   = bias[colBase + m];
    const int   half = lane >> 4;
#pragma unroll
    for (int r = 0; r < 8; ++r) {
        C[(size_t)(rowBase + r + 8 * half) * DOUT + colBase + m] = c[r] + bn;
    }
}

// ---------------------------------------------------------------------------
// Propagation: agg = dinv[i]^2 * h[i]  (self loop, no atomics)  then
//              agg[col] += dinv[row]*dinv[col]*h[row] per edge (atomic f32)
// ---------------------------------------------------------------------------
__global__ void prop_init_kernel(const float* __restrict__ h,
                                 const float* __restrict__ dinv,
                                 float* __restrict__ agg, int n) {
    int t = blockIdx.x * blockDim.x + threadIdx.x;   // n*32 threads
    if (t >= n * 32) return;
    int row = t >> 5, s = t & 31;
    float d2 = dinv[row] * dinv[row];
    float4 v = *(const float4*)(h + (size_t)row * D_FEAT + s * 4);
    v.x *= d2; v.y *= d2; v.z *= d2; v.w *= d2;
    *(float4*)(agg + (size_t)row * D_FEAT + s * 4) = v;
}

__global__ void prop_edges_kernel(const int* __restrict__ rows,
                                  const int* __restrict__ cols,
                                  const float* __restrict__ dinv,
                                  const float* __restrict__ h,
                                  float* __restrict__ agg, int nE) {
    const int lane       = threadIdx.x & 31;
    const int wave       = (blockIdx.x * blockDim.x + threadIdx.x) >> 5;
    const int totalWaves = (gridDim.x * blockDim.x) >> 5;

    for (int e = wave; e < nE; e += totalWaves) {
        int   r  = rows[e];
        int   cc = cols[e];
        float w  = dinv[r] * dinv[cc];
        float4 hv = *(const float4*)(h + (size_t)r * D_FEAT + lane * 4);
        float* dst = agg + (size_t)cc * D_FEAT + lane * 4;
        (void)__hip_atomic_fetch_add(dst + 0, w * hv.x, __ATOMIC_RELAXED, __HIP_MEMORY_SCOPE_AGENT);
        (void)__hip_atomic_fetch_add(dst + 1, w * hv.y, __ATOMIC_RELAXED, __HIP_MEMORY_SCOPE_AGENT);
        (void)__hip_atomic_fetch_add(dst + 2, w * hv.z, __ATOMIC_RELAXED, __HIP_MEMORY_SCOPE_AGENT);
        (void)__hip_atomic_fetch_add(dst + 3, w * hv.w, __ATOMIC_RELAXED, __HIP_MEMORY_SCOPE_AGENT);
        // latency-hiding: prefetch next edge's gather row (global_prefetch_b8)
        int e2 = e + totalWaves;
        if (e2 < nE) {
            int r2 = rows[e2];
            __builtin_prefetch(h + (size_t)r2 * D_FEAT + lane * 4, 0, 1);
        }
    }
}

// ---------------------------------------------------------------------------
// Row-wise L2 normalize (eps=1e-12) + ReLU; one wave32 per row.
// ---------------------------------------------------------------------------
__global__ void l2norm_relu_kernel(float* __restrict__ h, int n) {
    const int lane = threadIdx.x & 31;
    const int row  = (blockIdx.x * blockDim.x + threadIdx.x) >> 5;
    if (row >= n) return;
    float4 v = *(const float4*)(h + (size_t)row * D_FEAT + lane * 4);
    float s = v.x * v.x + v.y * v.y + v.z * v.z + v.w * v.w;
#pragma unroll
    for (int off = 16; off > 0; off >>= 1) s += __shfl_xor(s, off, 32);
    float nn = sqrtf(s);
    float sc = 1.0f / fmaxf(nn, 1e-12f);
    v.x = fmaxf(v.x * sc, 0.0f);
    v.y = fmaxf(v.y * sc, 0.0f);
    v.z = fmaxf(v.z * sc, 0.0f);
    v.w = fmaxf(v.w * sc, 0.0f);
    *(float4*)(h + (size_t)row * D_FEAT + lane * 4) = v;
}

// ---------------------------------------------------------------------------
// Host launcher
// ---------------------------------------------------------------------------
static inline size_t align256(size_t x) { return (x + 255) & ~(size_t)255; }

extern "C" void kernel_launch(void* const* d_in, const int* in_sizes, int n_in,
                              void* d_out, int out_size, void* d_ws, size_t ws_size,
                              hipStream_t stream) {
    const float* x    = (const float*)d_in[0];
    const int*   eidx = (const int*)d_in[1];
    const float* W1   = (const float*)d_in[2];
    const float* b1   = (const float*)d_in[3];
    const float* W2   = (const float*)d_in[4];
    const float* b2   = (const float*)d_in[5];
    const float* W3   = (const float*)d_in[6];
    const float* b3   = (const float*)d_in[7];
    float*       out  = (float*)d_out;

    const int N = in_sizes[0] / D_FEAT;     // 100000 (multiple of 16)
    const int E = in_sizes[1] / 2;          // 1600000
    const int* rows = eidx;                 // edge_index[0] = sources
    const int* cols = eidx + E;             // edge_index[1] = targets

    // Workspace: [dinv: N f32][bufA: N*128 f32][bufB: N*128 f32]
    // degree counts alias bufB (dead before bufB's first real use).
    char*  ws   = (char*)d_ws;
    size_t off  = 0;
    float* dinv = (float*)(ws + off); off += align256((size_t)N * 4);
    float* bufA = (float*)(ws + off); off += align256((size_t)N * D_FEAT * 4);
    float* bufB = (float*)(ws + off);
    unsigned* degc = (unsigned*)bufB;

    const int T = 256;
    dim3 blk(T);

    // degrees -> dinv
    init_deg_kernel<<<dim3((N + T - 1) / T), blk, 0, stream>>>(degc, N);
    count_edges_kernel<<<dim3((E + T - 1) / T), blk, 0, stream>>>(cols, degc, E);
    make_dinv_kernel<<<dim3((N + T - 1) / T), blk, 0, stream>>>(degc, dinv, N);

    const int gemmGrid = N / 16;            // 6250
    const int vecGrid  = (N * 32) / T;      // 12500 (exact)
    const int edgeGrid = 8192;              // grid-stride over edges

    // layer 0: x @ W1 + b1 -> prop -> l2norm+relu
    gemm_bias_wmma<128><<<dim3(gemmGrid), dim3(256), 0, stream>>>(x, W1, b1, bufA);
    prop_init_kernel<<<dim3(vecGrid), blk, 0, stream>>>(bufA, dinv, bufB, N);
    prop_edges_kernel<<<dim3(edgeGrid), blk, 0, stream>>>(rows, cols, dinv, bufA, bufB, E);
    l2norm_relu_kernel<<<dim3(vecGrid), blk, 0, stream>>>(bufB, N);

    // layer 1
    gemm_bias_wmma<128><<<dim3(gemmGrid), dim3(256), 0, stream>>>(bufB, W2, b2, bufA);
    prop_init_kernel<<<dim3(vecGrid), blk, 0, stream>>>(bufA, dinv, bufB, N);
    prop_edges_kernel<<<dim3(edgeGrid), blk, 0, stream>>>(rows, cols, dinv, bufA, bufB, E);
    l2norm_relu_kernel<<<dim3(vecGrid), blk, 0, stream>>>(bufB, N);

    // final: h @ W3 + b3 -> d_out (N x 64)
    gemm_bias_wmma<64><<<dim3(gemmGrid), dim3(128), 0, stream>>>(bufB, W3, b3, out);

    (void)n_in; (void)out_size; (void)ws_size;
}